// HierarchicalModel_39273180955181
// MI455X (gfx1250) — compile-verified
//
#include <hip/hip_runtime.h>
#include <math.h>

typedef float v2f __attribute__((ext_vector_type(2)));
typedef float v8f __attribute__((ext_vector_type(8)));

#define BB 2048
#define DD 3
#define KK 64
#define HH 768

// ---------------------------------------------------------------------------
// Kernel 1: per-sample gathered gemv on the WMMA pipe.
//   block  = 192 threads (6 waves) = one sample b
//   wave w = (d = w>>1, khalf = w&1)  -> computes logits[b, d, khalf*32 .. +31]
// Per wave: two 16-k tiles, accumulated over H with V_WMMA_F32_16X16X4_F32.
//   A tile: 16(k) x 4(h) of W[n]; lane l<16 holds row M=l, K={0,1};
//           lanes 16-31 hold row M=l-16, K={2,3}. We map the WMMA K axis to
//           h = {h0, h0+1, h0+4, h0+5} (wmma#0) and {h0+2,h0+3,h0+6,h0+7}
//           (wmma#1) so one float4 global load per lane feeds both wmmas and
//           the 32-lane load covers a contiguous 16x8 f32 tile of W (512B).
//   B tile: broadcast of pooled values (all 16 columns identical), built from
//           a lane-swizzled float4 LDS read with the same h mapping, so every
//           column of C holds the gemv result.
// ---------------------------------------------------------------------------
__global__ __launch_bounds__(192) void hier_logits_wmma(
    const float* __restrict__ pooled,   // [B, H]
    const int*   __restrict__ nodes,    // [B, D]
    const float* __restrict__ W,        // [N, K, H]
    const float* __restrict__ bias,     // [N, K]
    float*       __restrict__ out_logits) // [B, D, K]
{
  __shared__ float sp[HH];

  const int b = blockIdx.x;
  const int t = threadIdx.x;            // 0..191

  // Stage pooled[b,:] into LDS: 192 threads x float4 = 768 floats (one pass).
  ((float4*)sp)[t] = ((const float4*)(pooled + (size_t)b * HH))[t];
  __syncthreads();

  const int wave  = t >> 5;             // 0..5 (wave32)
  const int lane  = t & 31;
  const int d     = wave >> 1;          // depth 0..2
  const int khalf = wave & 1;           // which 32-k half
  const int n     = nodes[b * DD + d];

  const float* Wn = W + (size_t)n * (KK * HH);
  const int hi = lane >> 4;             // half-wave: selects K={0,1} vs {2,3}
  const int m  = lane & 15;             // row (k) within the 16-k tile

  const int k0 = khalf * 32;            // first 16-k tile
  const int k1 = k0 + 16;               // second 16-k tile

  // Fold the bias into the C initializer. Lane's VGPR i of tile T corresponds
  // to k = kT + 8*hi + i, identical across the 16 columns -> columns uniform.
  const float4 b00 = *(const float4*)(bias + n * KK + k0 + 8 * hi);
  const float4 b01 = *(const float4*)(bias + n * KK + k0 + 8 * hi + 4);
  const float4 b10 = *(const float4*)(bias + n * KK + k1 + 8 * hi);
  const float4 b11 = *(const float4*)(bias + n * KK + k1 + 8 * hi + 4);

  v8f acc0, acc1;
  acc0[0] = b00.x; acc0[1] = b00.y; acc0[2] = b00.z; acc0[3] = b00.w;
  acc0[4] = b01.x; acc0[5] = b01.y; acc0[6] = b01.z; acc0[7] = b01.w;
  acc1[0] = b10.x; acc1[1] = b10.y; acc1[2] = b10.z; acc1[3] = b10.w;
  acc1[4] = b11.x; acc1[5] = b11.y; acc1[6] = b11.z; acc1[7] = b11.w;

  const float* rowA0 = Wn + (size_t)(k0 + m) * HH + 4 * hi;
  const float* rowA1 = Wn + (size_t)(k1 + m) * HH + 4 * hi;
  const float* spl   = sp + 4 * hi;

#pragma unroll 4
  for (int h0 = 0; h0 < HH; h0 += 8) {
    // B operands: pooled broadcast, lane-swizzled (lanes<16: h0..h0+3,
    // lanes>=16: h0+4..h0+7) -> one ds_load_b128 feeds both wmmas.
    const float4 pv = *(const float4*)(spl + h0);
    // A operands: coalesced global_load_b128 per k-tile (16x8 f32 tile/wave).
    const float4 a0 = *(const float4*)(rowA0 + h0);
    const float4 a1 = *(const float4*)(rowA1 + h0);

    v2f bEven; bEven[0] = pv.x; bEven[1] = pv.y;   // h = {h0,h0+1 | h0+4,h0+5}
    v2f bOdd;  bOdd[0]  = pv.z; bOdd[1]  = pv.w;   // h = {h0+2,h0+3 | h0+6,h0+7}

    v2f aE0; aE0[0] = a0.x; aE0[1] = a0.y;
    v2f aO0; aO0[0] = a0.z; aO0[1] = a0.w;
    v2f aE1; aE1[0] = a1.x; aE1[1] = a1.y;
    v2f aO1; aO1[0] = a1.z; aO1[1] = a1.w;

    acc0 = __builtin_amdgcn_wmma_f32_16x16x4_f32(false, aE0, false, bEven,
                                                 (short)0, acc0, false, false);
    acc0 = __builtin_amdgcn_wmma_f32_16x16x4_f32(false, aO0, false, bOdd,
                                                 (short)0, acc0, false, false);
    acc1 = __builtin_amdgcn_wmma_f32_16x16x4_f32(false, aE1, false, bEven,
                                                 (short)0, acc1, false, false);
    acc1 = __builtin_amdgcn_wmma_f32_16x16x4_f32(false, aO1, false, bOdd,
                                                 (short)0, acc1, false, false);
  }

  // Every column of C is identical; let column-0 lane of each half-wave store
  // its 8 k-values per tile (k = kT + 8*hi + i).
  if (m == 0) {
    float* outp = out_logits + (size_t)(b * DD + d) * KK + 8 * hi;
    *(float4*)(outp + k0)     = make_float4(acc0[0], acc0[1], acc0[2], acc0[3]);
    *(float4*)(outp + k0 + 4) = make_float4(acc0[4], acc0[5], acc0[6], acc0[7]);
    *(float4*)(outp + k1)     = make_float4(acc1[0], acc1[1], acc1[2], acc1[3]);
    *(float4*)(outp + k1 + 4) = make_float4(acc1[4], acc1[5], acc1[6], acc1[7]);
  }
}

// ---------------------------------------------------------------------------
// Kernel 2: per-sample cross-entropy over the path. Tiny (1.5 MB of reads).
// ---------------------------------------------------------------------------
__global__ void hier_loss(const int* __restrict__ path,     // [B, D]
                          const float* __restrict__ logits, // [B, D, K]
                          float* __restrict__ losses)       // [B]
{
  const int b = blockIdx.x * blockDim.x + threadIdx.x;
  if (b >= BB) return;
  float loss = 0.0f;
#pragma unroll
  for (int d = 0; d < DD; ++d) {
    const float* base = logits + (size_t)(b * DD + d) * KK;
    float mx = -INFINITY;
    for (int k = 0; k < KK; ++k) mx = fmaxf(mx, base[k]);
    float s = 0.0f;
    for (int k = 0; k < KK; ++k) s += __expf(base[k] - mx);
    const float lse = mx + __logf(s);
    loss += lse - base[path[b * DD + d]];
  }
  losses[b] = loss;
}

extern "C" void kernel_launch(void* const* d_in, const int* in_sizes, int n_in,
                              void* d_out, int out_size, void* d_ws, size_t ws_size,
                              hipStream_t stream) {
  (void)in_sizes; (void)n_in; (void)out_size; (void)d_ws; (void)ws_size;
  const float* pooled = (const float*)d_in[0];   // [B,H] f32
  const int*   nodes  = (const int*)d_in[1];     // [B,D] i32
  const int*   path   = (const int*)d_in[2];     // [B,D] i32
  const float* W      = (const float*)d_in[3];   // [N,K,H] f32
  const float* bias   = (const float*)d_in[4];   // [N,K] f32

  float* out    = (float*)d_out;
  float* losses = out;           // [B]
  float* logits = out + BB;      // [B,D,K]

  hier_logits_wmma<<<BB, 192, 0, stream>>>(pooled, nodes, W, bias, logits);
  hier_loss<<<(BB + 255) / 256, 256, 0, stream>>>(path, logits, losses);
}